// DTWFeatures_85529978732672
// MI455X (gfx1250) — compile-verified
//
#include <hip/hip_runtime.h>
#include <math.h>

// ---------------------------------------------------------------------------
// DTW features on MI455X (gfx1250, wave32).
//
//   x:     (B=64, d=3, T=1024) f32
//   patts: (P=32, d=3, L=32)  f32
//   out:   (B=64, P=32, T=1024) f32,  out[b,p,t] = D[L-1, t] of the weighted
//          min-plus DTW DP with w = 0.1^(1/32).
//
// One wave32 per (b,p): lane j holds DP state D[j, t].
//  * dist cross-term via V_WMMA_F32_16X16X4_F32: A row (t) = [x0,x1,x2,|x|^2],
//    B col (l) = [-2q0,-2q1,-2q2,1] -> tile = |x|^2 - 2 x.q; +|q|^2 and the
//    fast v_sqrt_f32 are applied to the WMMA registers before the LDS scatter.
//  * Per timestep, D[j] = d[j] + min(D[j-1], M[j]) is an affine min-plus map
//    f_j(x) = min(x + d[j], M[j] + d[j]).  A 5-step pair scan over (p,q)
//    composites solves the column:
//        (p,q)_new = (p_up + p, min(q_up + p, q)),   D[j] = q_j.
//    Raw ds_bpermute with a shared precomputed index per step lets both
//    shuffles issue back-to-back under one s_wait_dscnt.
// ---------------------------------------------------------------------------

typedef __attribute__((ext_vector_type(2))) float v2f;
typedef __attribute__((ext_vector_type(8))) float v8f;

#define B_N 64
#define P_N 32
#define L_N 32
#define T_N 1024
#define DTW_W 0.93057205f       /* 0.1 ** (1/32) */
#define BIGF  3.0e38f
#define TILE_STRIDE 33          /* 32 + 1 pad: kills LDS bank conflicts */
#define WAVE_LDS (16 * TILE_STRIDE + 16)

__device__ __forceinline__ float bperm_f(int byte_idx, float v) {
  return __int_as_float(__builtin_amdgcn_ds_bpermute(byte_idx, __float_as_int(v)));
}

__global__ __launch_bounds__(256) void dtw_wmma_kernel(
    const float* __restrict__ x,
    const float* __restrict__ patts,
    float* __restrict__ out) {
  const int lane = threadIdx.x & 31;
  const int wib  = threadIdx.x >> 5;               // wave in block: 0..7
  const int wave = blockIdx.x * 8 + wib;           // 0..2047
  const int b = wave >> 5;                         // batch
  const int p = wave & 31;                         // pattern

  __shared__ float lds[8 * WAVE_LDS];
  float* tile   = &lds[wib * WAVE_LDS];            // 16 x 32 (stride 33) dist tile
  float* ostage = tile + 16 * TILE_STRIDE;         // 16 staged outputs

  const float* xb = x + b * 3 * T_N;               // x[b, d, t]
  const float* pp = patts + p * 3 * L_N;           // patts[p, d, l]

  const int  nn = lane & 15;
  const bool hi = lane >= 16;

  // Precomputed bpermute byte-indices (wrap via addr bits [6:2] is harmless --
  // results for lane < off are masked) and lane predicates for scan masking.
  const int im1 = (lane - 1)  << 2;
  const int i1  = im1;
  const int i2  = (lane - 2)  << 2;
  const int i4  = (lane - 4)  << 2;
  const int i8  = (lane - 8)  << 2;
  const int i16 = (lane - 16) << 2;
  const bool g1 = lane >= 1, g2 = lane >= 2, g4 = lane >= 4,
             g8 = lane >= 8, g16 = lane >= 16;

  // B-matrix (4 x 16) registers for the two l-blocks + per-column |q|^2.
  // 32-bit B layout: VGPR0 lanes0-15 = K0 row, lanes16-31 = K2 row;
  //                  VGPR1 lanes0-15 = K1 row, lanes16-31 = K3 row.
  v2f bm0, bm1;
  float p2lo, p2hi;
  {
    const float a0 = pp[0 * L_N + nn];
    const float a1 = pp[1 * L_N + nn];
    const float a2 = pp[2 * L_N + nn];
    p2lo = a0 * a0 + a1 * a1 + a2 * a2;            // |q(l=nn)|^2
    bm0.x = hi ? (-2.0f * a2) : (-2.0f * a0);      // K2 : K0
    bm0.y = hi ? 1.0f         : (-2.0f * a1);      // K3(ones) : K1
    const float c0 = pp[0 * L_N + 16 + nn];
    const float c1 = pp[1 * L_N + 16 + nn];
    const float c2 = pp[2 * L_N + 16 + nn];
    p2hi = c0 * c0 + c1 * c1 + c2 * c2;            // |q(l=16+nn)|^2
    bm1.x = hi ? (-2.0f * c2) : (-2.0f * c0);
    bm1.y = hi ? 1.0f         : (-2.0f * c1);
  }

  const float w = DTW_W;
  float cprev = 0.0f;                   // D[lane, t-1]

  // One pair-scan step: combine with composite from OFF lanes up.
  // new_q uses pre-update p_self; identity upstream = (0, +inf).
#define SCAN_STEP(IDX, G)                            \
  {                                                  \
    float pu = bperm_f((IDX), pS);                   \
    float qu = bperm_f((IDX), qS);                   \
    pu = (G) ? pu : 0.0f;                            \
    qu = (G) ? qu : BIGF;                            \
    qS = fminf(qu + pS, qS);                         \
    pS = pu + pS;                                    \
  }

  for (int blk = 0; blk < T_N / 16; ++blk) {
    const int t0 = blk * 16;

    // Prefetch next x tile into cache (global_prefetch_b8).
    if (blk < T_N / 16 - 1) {
      __builtin_prefetch(xb + t0 + 16, 0, 3);
      __builtin_prefetch(xb + 2 * T_N + t0 + 16, 0, 3);
    }

    // A-matrix (16 x 4) registers: row M = local timestep.
    // 32-bit A layout: VGPR0 lanes0-15 = K0, lanes16-31 = K2;
    //                  VGPR1 lanes0-15 = K1, lanes16-31 = K3.
    const int tm = t0 + nn;
    const float x0  = xb[0 * T_N + tm];
    const float x1  = xb[1 * T_N + tm];
    const float x2c = xb[2 * T_N + tm];
    const float xn  = x0 * x0 + x1 * x1 + x2c * x2c;   // |x|^2 in K3 slot
    v2f am;
    am.x = hi ? x2c : x0;
    am.y = hi ? xn  : x1;

    // xp tile = A x B : 16 timesteps x 16 pattern positions, twice for l=0..31.
    v8f acc = {};
    v8f dlo = __builtin_amdgcn_wmma_f32_16x16x4_f32(
        false, am, false, bm0, (short)0, acc, false, false);
    v8f dhi = __builtin_amdgcn_wmma_f32_16x16x4_f32(
        false, am, false, bm1, (short)0, acc, false, false);

    // Finish the distance (add |q|^2, clamp, fast sqrt) on the WMMA output
    // registers -- 16 pipelined TRANS sqrts, off the DP chain -- then scatter
    // to LDS in (t_local, l) layout.
    // C/D layout: VGPR k -> lanes0-15: (M=k, N=lane); lanes16-31: (M=k+8, N=lane-16).
    const int row0 = hi ? 8 : 0;
#pragma unroll
    for (int k = 0; k < 8; ++k) {
      const float dl = __builtin_amdgcn_sqrtf(fmaxf(dlo[k] + p2lo, 1e-12f));
      const float dh = __builtin_amdgcn_sqrtf(fmaxf(dhi[k] + p2hi, 1e-12f));
      tile[(row0 + k) * TILE_STRIDE + nn]      = dl;
      tile[(row0 + k) * TILE_STRIDE + nn + 16] = dh;
    }

    // Sequential DP over the 16 timesteps of this tile.
#pragma unroll 4
    for (int k = 0; k < 16; ++k) {
      const float d = tile[k * TILE_STRIDE + lane];

      float c;
      if (blk == 0 && k == 0) {
        // First column: D[j,0] = cumsum(dist[:,0]) -- plain prefix sum.
        float S = d;
        {
          float v;
          v = bperm_f(i1,  S); S += g1  ? v : 0.0f;
          v = bperm_f(i2,  S); S += g2  ? v : 0.0f;
          v = bperm_f(i4,  S); S += g4  ? v : 0.0f;
          v = bperm_f(i8,  S); S += g8  ? v : 0.0f;
          v = bperm_f(i16, S); S += g16 ? v : 0.0f;
        }
        c = S;
      } else {
        // M[j] = w*min(Dprev[j], Dprev[j-1]); lane0's wrapped shuffle value is
        // masked to its own value so M[0] = w*Dprev[0].
        float cm1 = bperm_f(im1, cprev);
        cm1 = g1 ? cm1 : cprev;
        const float m = w * fminf(cprev, cm1);

        // Min-plus affine pair scan: f_j(x) = min(x + pS, qS).
        float pS = d;
        float qS = m + d;
        SCAN_STEP(i1,  g1)
        SCAN_STEP(i2,  g2)
        SCAN_STEP(i4,  g4)
        SCAN_STEP(i8,  g8)
        SCAN_STEP(i16, g16)
        c = qS;                                   // D[j] = F_j(+inf)
      }
      cprev = c;
      if (lane == 31) ostage[k] = c;             // D[L-1, t] is the feature
    }

    // Coalesced 16-wide output store for this t tile.
    const float oval = ostage[lane & 15];
    if (lane < 16) out[(b * P_N + p) * T_N + t0 + lane] = oval;
  }
#undef SCAN_STEP
}

extern "C" void kernel_launch(void* const* d_in, const int* in_sizes, int n_in,
                              void* d_out, int out_size, void* d_ws, size_t ws_size,
                              hipStream_t stream) {
  (void)in_sizes; (void)n_in; (void)out_size; (void)d_ws; (void)ws_size;
  const float* x     = (const float*)d_in[0];
  const float* patts = (const float*)d_in[1];
  float* out = (float*)d_out;

  // 2048 (b,p) waves / 8 waves per 256-thread block = 256 blocks.
  dim3 grid(256), block(256);
  hipLaunchKernelGGL(dtw_wmma_kernel, grid, block, 0, stream, x, patts, out);
}